// QuadratureFunction_71244917506065
// MI455X (gfx1250) — compile-verified
//
#include <hip/hip_runtime.h>

typedef float v2f __attribute__((ext_vector_type(2)));
typedef float v4f __attribute__((ext_vector_type(4)));
typedef float v8f __attribute__((ext_vector_type(8)));

#define NBASIS 10
#define NQ 16

// One wave32 per cell. Each cell's 1x10 (dofs) times 10x16 (y) product is done
// with three V_WMMA_F32_16X16X4_F32 accumulations (K padded 10 -> 12):
//   A[m=q, k] = y[c, k0+k, q]   (A f32 16x4 layout: lanes 0-15 K=k0,k0+1 in
//                                vgpr0/1; lanes 16-31 K=k0+2,k0+3)
//   B[k, n]   = dof[k0+k]       (constant across N)
//   D[m, n]   = out[c, m]       (identical in every column n)
// D layout: lane<16 vgpr r = (M=r, N=lane); lane>=16 vgpr r = (M=8+r).
// So lane 0 holds out[c,0..7] and lane 16 holds out[c,8..15].
__global__ __launch_bounds__(256) void quadfn_wmma_kernel(
    const float* __restrict__ vertex_dofs,   // [V,1]
    const float* __restrict__ edge_dofs,     // [E,2]
    const float* __restrict__ face_dofs,     // [C,1]
    const float* __restrict__ y,             // [C,10,16]
    const int*   __restrict__ faces,         // [C,3]
    const int*   __restrict__ f2e,           // [C,3]
    const int*   __restrict__ ori,           // [C,3]
    float*       __restrict__ out,           // [C,16]
    int num_cells)
{
    const int wave = (int)((blockIdx.x * blockDim.x + threadIdx.x) >> 5);
    if (wave >= num_cells) return;              // wave-uniform exit: EXEC all-1s below
    const int c    = __builtin_amdgcn_readfirstlane(wave);  // force SGPR -> scalar gathers
    const int lane = (int)(threadIdx.x & 31u);
    const int half = lane >> 4;                 // 0: lanes 0-15, 1: lanes 16-31
    const int q    = lane & 15;                 // quadrature point = matrix row M

    // ---- uniform topology + dof gathers (scalar loads) -------------------
    const int vi0 = faces[3 * c + 0];
    const int vi1 = faces[3 * c + 1];
    const int vi2 = faces[3 * c + 2];
    const float d0 = vertex_dofs[vi0];
    const float d1 = vertex_dofs[vi1];
    const float d2 = vertex_dofs[vi2];

    const int e0 = f2e[3 * c + 0], o0 = ori[3 * c + 0];
    const int e1 = f2e[3 * c + 1], o1 = ori[3 * c + 1];
    const int e2 = f2e[3 * c + 2], o2 = ori[3 * c + 2];
    const float e0a = edge_dofs[2 * e0], e0b = edge_dofs[2 * e0 + 1];
    const float e1a = edge_dofs[2 * e1], e1b = edge_dofs[2 * e1 + 1];
    const float e2a = edge_dofs[2 * e2], e2b = edge_dofs[2 * e2 + 1];
    // orientation==1: keep [a,b]; orientation==0: flipped [b,a]
    const float d3 = o0 ? e0a : e0b;
    const float d4 = o0 ? e0b : e0a;
    const float d5 = o1 ? e1a : e1b;
    const float d6 = o1 ? e1b : e1a;
    const float d7 = o2 ? e2a : e2b;
    const float d8 = o2 ? e2b : e2a;
    const float d9 = face_dofs[c];

    // ---- per-lane y loads (coalesced: lane stride 4B within a b-row) -----
    const float* yb = y + (size_t)c * (NBASIS * NQ) + q;  // yb[b*16] = y[c,b,q]
    const int bA = half << 1;  // this lane-half's first K within each chunk

    v2f A0, B0, A1, B1, A2, B2;
    // chunk 0: K = 0..3  (dofs d0..d3)
    A0.x = yb[(0 + bA) * NQ];
    A0.y = yb[(1 + bA) * NQ];
    B0.x = half ? d2 : d0;
    B0.y = half ? d3 : d1;
    // chunk 1: K = 4..7  (dofs d4..d7)
    A1.x = yb[(4 + bA) * NQ];
    A1.y = yb[(5 + bA) * NQ];
    B1.x = half ? d6 : d4;
    B1.y = half ? d7 : d5;
    // chunk 2: K = 8..11 (dofs d8,d9, then zero padding; loads stay in-bounds)
    A2.x = half ? 0.0f : yb[8 * NQ];
    A2.y = half ? 0.0f : yb[9 * NQ];
    B2.x = half ? 0.0f : d8;
    B2.y = half ? 0.0f : d9;

    // ---- 3x WMMA f32 16x16x4 accumulation chain --------------------------
    v8f acc = {};
    acc = __builtin_amdgcn_wmma_f32_16x16x4_f32(false, A0, false, B0,
                                                (short)0, acc, false, false);
    acc = __builtin_amdgcn_wmma_f32_16x16x4_f32(false, A1, false, B1,
                                                (short)0, acc, false, false);
    acc = __builtin_amdgcn_wmma_f32_16x16x4_f32(false, A2, false, B2,
                                                (short)0, acc, false, false);

    // ---- store: lane 0 -> out[c,0..7], lane 16 -> out[c,8..15] -----------
    if (q == 0) {
        v4f* p = (v4f*)(out + (size_t)c * NQ + (half << 3));
        v4f lo, hi;
        lo.x = acc[0]; lo.y = acc[1]; lo.z = acc[2]; lo.w = acc[3];
        hi.x = acc[4]; hi.y = acc[5]; hi.z = acc[6]; hi.w = acc[7];
        p[0] = lo;
        p[1] = hi;
    }
}

extern "C" void kernel_launch(void* const* d_in, const int* in_sizes, int n_in,
                              void* d_out, int out_size, void* d_ws, size_t ws_size,
                              hipStream_t stream) {
    const float* vertex_dofs = (const float*)d_in[0];
    const float* edge_dofs   = (const float*)d_in[1];
    const float* face_dofs   = (const float*)d_in[2];
    const float* y           = (const float*)d_in[3];
    const int*   faces       = (const int*)d_in[4];
    const int*   f2e         = (const int*)d_in[5];
    const int*   ori         = (const int*)d_in[6];
    float* out = (float*)d_out;

    const int num_cells = in_sizes[2];  // face_dofs is [C,1]

    const int threads = 256;            // 8 waves/block -> 8 cells/block
    const long long total_threads = (long long)num_cells * 32;
    const int blocks = (int)((total_threads + threads - 1) / threads);

    quadfn_wmma_kernel<<<blocks, threads, 0, stream>>>(
        vertex_dofs, edge_dofs, face_dofs, y, faces, f2e, ori, out, num_cells);
}